// Net_62371515072862
// MI455X (gfx1250) — compile-verified
//
#include <hip/hip_runtime.h>

// ---------------------------------------------------------------------------
// 2-layer GCN link predictor for MI455X (gfx1250, wave32).
// GEMMs use V_WMMA_F32_16X16X4_F32 (full f32 accuracy, matches reference).
// Scatter phase is the memory-bound hot spot (~2.2 GB random traffic) and
// uses hardware global_atomic_add_f32 with float4/float2 coalesced gathers.
// ---------------------------------------------------------------------------

typedef __attribute__((ext_vector_type(2))) float v2f;
typedef __attribute__((ext_vector_type(8))) float v8f;

__device__ __forceinline__ v8f wmma_f32_16x16x4(v2f a, v2f b, v8f c) {
  // 8-arg pattern: (neg_a, A, neg_b, B, c_mod, C, reuse_a, reuse_b)
  return __builtin_amdgcn_wmma_f32_16x16x4_f32(false, a, false, b, (short)0, c,
                                               false, false);
}

__device__ __forceinline__ void atomic_add_f32(float* p, float v) {
  __hip_atomic_fetch_add(p, v, __ATOMIC_RELAXED, __HIP_MEMORY_SCOPE_AGENT);
}

// ---------------- degree / normalization ----------------------------------

__global__ void k_init_deg(float* __restrict__ deg, int n) {
  int i = blockIdx.x * blockDim.x + threadIdx.x;
  if (i < n) deg[i] = 1.0f;  // self-loop contributes 1 to every node
}

__global__ void k_count_deg(const long long* __restrict__ dst,
                            float* __restrict__ deg, int nE) {
  int i = blockIdx.x * blockDim.x + threadIdx.x;
  if (i < nE) atomic_add_f32(&deg[dst[i]], 1.0f);
}

__global__ void k_deg_to_dinv(float* __restrict__ deg, int n) {
  int i = blockIdx.x * blockDim.x + threadIdx.x;
  if (i < n) deg[i] = rsqrtf(deg[i]);  // deg >= 1 always (self-loops)
}

// ---------------- WMMA GEMM: C[M,NOUT] = A[M,K] @ B[K,NOUT] ---------------
// blockDim = (NOUT/16)*32 threads; one wave per 16-column tile.
// A tile (16 x K) staged in LDS with +4 float padding -> conflict-free
// ds_load_b64 fragment reads.  ISA 7.12.2 layouts:
//   A frag: a[v] = A[m][k0 + v + 2*half]     (m = lane&15, half = lane>>4)
//   B frag: b[v] = B[k0 + v + 2*half][n]
//   D:      d[v] -> row (v + 8*half), col n

template <int K, int NOUT>
__global__ void k_gemm_wmma(const float* __restrict__ A,
                            const float* __restrict__ B,
                            float* __restrict__ C) {
  constexpr int LDSW = K + 4;
  __shared__ float As[16 * LDSW];

  const int tid = threadIdx.x;
  const int m0 = blockIdx.x * 16;

  // cooperative float4 load of the 16 x K A-tile
  constexpr int KV = K / 4;
  for (int idx = tid; idx < 16 * KV; idx += blockDim.x) {
    const int r = idx / KV, c4 = idx % KV;
    const float4 v =
        ((const float4*)(A + (size_t)(m0 + r) * K))[c4];
    float* d = &As[r * LDSW + c4 * 4];
    d[0] = v.x; d[1] = v.y; d[2] = v.z; d[3] = v.w;
  }
  __syncthreads();

  const int wave = tid >> 5;
  const int lane = tid & 31;
  const int half = lane >> 4;
  const int mn   = lane & 15;
  const int n0   = wave * 16;

  const float* __restrict__ Arow = &As[mn * LDSW + 2 * half];
  const float* __restrict__ Bcol = B + (size_t)(2 * half) * NOUT + n0 + mn;

  v8f acc0 = {};
  v8f acc1 = {};
  for (int k0 = 0; k0 < K; k0 += 8) {
    const float2 av0 = *(const float2*)(Arow + k0);
    const float2 av1 = *(const float2*)(Arow + k0 + 4);
    v2f a0; a0.x = av0.x; a0.y = av0.y;
    v2f a1; a1.x = av1.x; a1.y = av1.y;
    v2f b0; b0.x = Bcol[(size_t)k0 * NOUT];
            b0.y = Bcol[(size_t)(k0 + 1) * NOUT];
    v2f b1; b1.x = Bcol[(size_t)(k0 + 4) * NOUT];
            b1.y = Bcol[(size_t)(k0 + 5) * NOUT];
    acc0 = wmma_f32_16x16x4(a0, b0, acc0);
    acc1 = wmma_f32_16x16x4(a1, b1, acc1);
  }

  float* __restrict__ Crow = C + (size_t)m0 * NOUT + n0 + mn;
#pragma unroll
  for (int v = 0; v < 8; ++v)
    Crow[(size_t)(v + 8 * half) * NOUT] = acc0[v] + acc1[v];
}

// ---------------- edge scatter: agg[dst] += h[src] * dinv[s]*dinv[d] ------
// One wave32 per edge; each lane handles DV contiguous floats (D = 32*DV).

template <int DV>
__global__ void k_scatter(const float* __restrict__ h,
                          const long long* __restrict__ src,
                          const long long* __restrict__ dst,
                          const float* __restrict__ dinv,
                          float* __restrict__ agg, int nE) {
  const int e = (int)((blockIdx.x * blockDim.x + threadIdx.x) >> 5);
  const int lane = threadIdx.x & 31;
  if (e >= nE) return;
  const long long s = src[e], d = dst[e];
  const float norm = dinv[s] * dinv[d];
  constexpr int D = 32 * DV;
  const float* __restrict__ hp = h + (size_t)s * D + lane * DV;
  float* __restrict__ op = agg + (size_t)d * D + lane * DV;
  if constexpr (DV == 4) {
    const float4 v = *(const float4*)hp;
    atomic_add_f32(op + 0, v.x * norm);
    atomic_add_f32(op + 1, v.y * norm);
    atomic_add_f32(op + 2, v.z * norm);
    atomic_add_f32(op + 3, v.w * norm);
  } else {
    const float2 v = *(const float2*)hp;
    atomic_add_f32(op + 0, v.x * norm);
    atomic_add_f32(op + 1, v.y * norm);
  }
}

// ---------------- self-loop + bias (+ReLU) fusion -------------------------
// inout[i] = (RELU?) relu( agg[i] + h[i]*dinv[node]^2 + bias[col] )

template <bool RELU, int D>
__global__ void k_finish(float* __restrict__ inout,
                         const float* __restrict__ h,
                         const float* __restrict__ dinv,
                         const float* __restrict__ bias, int total) {
  const int i = blockIdx.x * blockDim.x + threadIdx.x;
  if (i >= total) return;
  const int node = i / D;
  const int c = i & (D - 1);
  const float s = dinv[node];
  float v = inout[i] + h[i] * (s * s) + bias[c];
  inout[i] = RELU ? fmaxf(v, 0.0f) : v;
}

// ---------------- decoder: out[e] = dot(z[src], z[dst]) over 64 dims ------

__global__ void k_edge_dot(const float* __restrict__ z,
                           const long long* __restrict__ src,
                           const long long* __restrict__ dst,
                           float* __restrict__ out, int nE) {
  const int e = (int)((blockIdx.x * blockDim.x + threadIdx.x) >> 5);
  const int lane = threadIdx.x & 31;
  if (e >= nE) return;
  const long long s = src[e], d = dst[e];
  const float2 a = *(const float2*)(z + (size_t)s * 64 + lane * 2);
  const float2 b = *(const float2*)(z + (size_t)d * 64 + lane * 2);
  float v = a.x * b.x + a.y * b.y;
#pragma unroll
  for (int off = 16; off > 0; off >>= 1) v += __shfl_xor(v, off, 32);
  if (lane == 0) out[e] = v;
}

// ---------------------------------------------------------------------------

extern "C" void kernel_launch(void* const* d_in, const int* in_sizes, int n_in,
                              void* d_out, int out_size, void* d_ws,
                              size_t ws_size, hipStream_t stream) {
  const float* x      = (const float*)d_in[0];       // [N,256]
  const long long* ei = (const long long*)d_in[1];   // [2,E]
  const long long* el = (const long long*)d_in[2];   // [2,EL]
  const float* W1     = (const float*)d_in[3];       // [256,128]
  const float* b1     = (const float*)d_in[4];       // [128]
  const float* W2     = (const float*)d_in[5];       // [128,64]
  const float* b2     = (const float*)d_in[6];       // [64]
  float* out          = (float*)d_out;               // [EL]

  const int N  = in_sizes[0] / 256;   // 100000 (divisible by 16)
  const int E  = in_sizes[1] / 2;     // 1600000
  const int EL = in_sizes[2] / 2;     // 500000

  const long long* eSrc = ei;
  const long long* eDst = ei + E;
  const long long* lSrc = el;
  const long long* lDst = el + EL;

  // workspace layout (floats), all 256B-aligned chunks
  float* ws   = (float*)d_ws;
  float* dinv = ws;                              // N
  float* xw1  = dinv + ((N + 63) & ~63);         // N*128
  float* agg1 = xw1 + (size_t)N * 128;           // N*128  (becomes h1)
  float* hw2  = agg1 + (size_t)N * 128;          // N*64
  float* agg2 = hw2 + (size_t)N * 64;            // N*64   (becomes z)

  const int T = 256;

  // 1) symmetric normalization coefficients
  k_init_deg<<<(N + T - 1) / T, T, 0, stream>>>(dinv, N);
  k_count_deg<<<(E + T - 1) / T, T, 0, stream>>>(eDst, dinv, E);
  k_deg_to_dinv<<<(N + T - 1) / T, T, 0, stream>>>(dinv, N);

  // 2) layer 1: h1 = relu( scatter(norm * (x@W1)) + selfloop + b1 )
  hipMemsetAsync(agg1, 0, (size_t)N * 128 * sizeof(float), stream);
  k_gemm_wmma<256, 128><<<N / 16, 256, 0, stream>>>(x, W1, xw1);
  {
    const int edgesPerBlock = T / 32;
    k_scatter<4><<<(E + edgesPerBlock - 1) / edgesPerBlock, T, 0, stream>>>(
        xw1, eSrc, eDst, dinv, agg1, E);
  }
  k_finish<true, 128><<<((size_t)N * 128 + T - 1) / T, T, 0, stream>>>(
      agg1, xw1, dinv, b1, N * 128);

  // 3) layer 2: z = scatter(norm * (h1@W2)) + selfloop + b2
  hipMemsetAsync(agg2, 0, (size_t)N * 64 * sizeof(float), stream);
  k_gemm_wmma<128, 64><<<N / 16, 128, 0, stream>>>(agg1, W2, hw2);
  {
    const int edgesPerBlock = T / 32;
    k_scatter<2><<<(E + edgesPerBlock - 1) / edgesPerBlock, T, 0, stream>>>(
        hw2, eSrc, eDst, dinv, agg2, E);
  }
  k_finish<false, 64><<<((size_t)N * 64 + T - 1) / T, T, 0, stream>>>(
      agg2, hw2, dinv, b2, N * 64);

  // 4) link decoder: dot(z[src], z[dst])
  {
    const int edgesPerBlock = T / 32;
    k_edge_dot<<<(EL + edgesPerBlock - 1) / edgesPerBlock, T, 0, stream>>>(
        agg2, lSrc, lDst, out, EL);
  }
}